// LightGCN_xij_item_personal_matrix_nohyper_50294067036544
// MI455X (gfx1250) — compile-verified
//
#include <hip/hip_runtime.h>
#include <hip/hip_bf16.h>
#include <math.h>

#define U_CNT   100000
#define I_CNT   50000
#define N_CNT   150000
#define D_DIM   64
#define X_DIM   16
#define NNZ_CNT 2400000
#define B_CNT   16384
#define NLAYERS 3

typedef __attribute__((ext_vector_type(2))) float v2f;
typedef __attribute__((ext_vector_type(8))) float v8f;

// x = acc = concat(emb_user, emb_item)
__global__ void lg_init_kernel(const float* __restrict__ eu, const float* __restrict__ ei,
                               float* __restrict__ x, float* __restrict__ acc) {
  long long idx = (long long)blockIdx.x * blockDim.x + threadIdx.x;
  const long long total = (long long)N_CNT * D_DIM;
  if (idx >= total) return;
  const long long usz = (long long)U_CNT * D_DIM;
  float v = (idx < usz) ? eu[idx] : ei[idx - usz];
  x[idx] = v;
  acc[idx] = v;
}

__global__ void lg_zero_kernel(float* __restrict__ p, long long n) {
  long long idx = (long long)blockIdx.x * blockDim.x + threadIdx.x;
  if (idx < n) p[idx] = 0.0f;
}

// one wave per edge: 32 lanes x float2 = 64-wide row; scatter via fp32 atomics (L2-resident)
__global__ void lg_spmm_kernel(const int* __restrict__ rows, const int* __restrict__ cols,
                               const float* __restrict__ vals,
                               const float* __restrict__ x, float* __restrict__ out) {
  const int lane = threadIdx.x & 31;
  const int e = blockIdx.x * (blockDim.x >> 5) + (threadIdx.x >> 5);
  if (e >= NNZ_CNT) return;
  const int r = rows[e];
  const int c = cols[e];
  const float v = vals[e];
  const float2 xv = ((const float2*)(x + (long long)c * D_DIM))[lane];
  float* dst = out + (long long)r * D_DIM + lane * 2;
  atomicAdd(dst, v * xv.x);
  atomicAdd(dst + 1, v * xv.y);
}

__global__ void lg_accadd_kernel(float* __restrict__ acc, const float* __restrict__ xn, long long n) {
  long long idx = (long long)blockIdx.x * blockDim.x + threadIdx.x;
  if (idx < n) acc[idx] += xn[idx];
}

// One wave per 16 batch rows. For each of user/item:
//   lin[16,64] = (0.25*acc[gathered 16 rows, 64]) @ W^T via V_WMMA_F32_16X16X4_F32
// A tile per k-step: lane L (m=L%16, hi=L/16): a = {A[m][4k+2hi], A[m][4k+2hi+1]}
// B tile:                                     b = {W[n0+m][4k+2hi], W[n0+m][4k+2hi+1]}  (B[k][n]=W[n][k])
// D: VGPR r, lane L -> out row r0 + r + 8*hi, col n0 + m
__global__ void lg_gemm_wmma_kernel(const int* __restrict__ users, const int* __restrict__ items,
                                    const float* __restrict__ acc,
                                    const float* __restrict__ w_user, const float* __restrict__ w_item,
                                    float* __restrict__ ulin, float* __restrict__ ilin) {
  const int lane = threadIdx.x & 31;
  const int m = lane & 15;
  const int hi = lane >> 4;
  const int r0 = blockIdx.x * 16;

#pragma unroll
  for (int which = 0; which < 2; ++which) {
    const int id = which == 0 ? users[r0 + m] : items[r0 + m];
    const float* arow = acc + ((long long)(which == 0 ? id : (U_CNT + id))) * D_DIM;
    const float* W = (which == 0) ? w_user : w_item;
    float* lin = (which == 0) ? ulin : ilin;

    v2f a[16];
#pragma unroll
    for (int ks = 0; ks < 16; ++ks) {
      float2 t = *(const float2*)(arow + ks * 4 + hi * 2);
      a[ks].x = t.x * 0.25f;   // light_out = acc / (n_layers+1)
      a[ks].y = t.y * 0.25f;
    }

#pragma unroll
    for (int nt = 0; nt < 4; ++nt) {
      v8f c = {};
      const float* wb = W + (long long)(nt * 16 + m) * D_DIM + hi * 2;
#pragma unroll
      for (int ks = 0; ks < 16; ++ks) {
        float2 t = *(const float2*)(wb + ks * 4);
        v2f b; b.x = t.x; b.y = t.y;
        c = __builtin_amdgcn_wmma_f32_16x16x4_f32(
            /*neg_a=*/false, a[ks], /*neg_b=*/false, b,
            /*c_mod=*/(short)0, c, /*reuse_a=*/false, /*reuse_b=*/false);
      }
#pragma unroll
      for (int r = 0; r < 8; ++r) {
        lin[(long long)(r0 + r + hi * 8) * D_DIM + nt * 16 + m] = c[r];
      }
    }
  }
}

// one wave per batch row: softmax(u[80]) . sigmoid(v[80])
__global__ void lg_gamma_kernel(const int* __restrict__ users, const int* __restrict__ items,
                                const unsigned char* __restrict__ xij,
                                const float* __restrict__ ulin, const float* __restrict__ ilin,
                                const float* __restrict__ exu,
                                const float* __restrict__ ei1, const float* __restrict__ ei0,
                                float* __restrict__ out) {
  const int lane = threadIdx.x & 31;
  const int b = blockIdx.x * (blockDim.x >> 5) + (threadIdx.x >> 5);
  if (b >= B_CNT) return;
  const int u = users[b];
  const int it = items[b];
  const float* vtab = (xij[b] != 0) ? (ei1 + (long long)it * X_DIM)
                                    : (ei0 + (long long)it * X_DIM);
  float up[3], vp[3];
  float mx = -1e30f;
#pragma unroll
  for (int s = 0; s < 3; ++s) {
    const int i = lane + s * 32;
    float uv, vv;
    if (i < 64)      { uv = ulin[(long long)b * 64 + i]; vv = ilin[(long long)b * 64 + i]; }
    else if (i < 80) { uv = exu[(long long)u * X_DIM + (i - 64)]; vv = vtab[i - 64]; }
    else             { uv = -1e30f; vv = 0.0f; }
    up[s] = uv; vp[s] = vv;
    mx = fmaxf(mx, uv);
  }
#pragma unroll
  for (int off = 16; off >= 1; off >>= 1) mx = fmaxf(mx, __shfl_xor(mx, off, 32));

  float ev[3];
  float se = 0.0f;
#pragma unroll
  for (int s = 0; s < 3; ++s) {
    const int i = lane + s * 32;
    const float e = (i < 80) ? expf(up[s] - mx) : 0.0f;
    ev[s] = e; se += e;
  }
#pragma unroll
  for (int off = 16; off >= 1; off >>= 1) se += __shfl_xor(se, off, 32);

  float dot = 0.0f;
#pragma unroll
  for (int s = 0; s < 3; ++s) {
    const int i = lane + s * 32;
    if (i < 80) {
      const float uu = ev[s] / se;
      const float sg = 1.0f / (1.0f + expf(-vp[s]));
      dot += uu * sg;
    }
  }
#pragma unroll
  for (int off = 16; off >= 1; off >>= 1) dot += __shfl_xor(dot, off, 32);
  if (lane == 0) out[b] = dot;
}

extern "C" void kernel_launch(void* const* d_in, const int* in_sizes, int n_in,
                              void* d_out, int out_size, void* d_ws, size_t ws_size,
                              hipStream_t stream) {
  (void)in_sizes; (void)n_in; (void)out_size; (void)ws_size;
  const int*   users = (const int*)d_in[0];
  const int*   items = (const int*)d_in[1];
  const unsigned char* xij = (const unsigned char*)d_in[2];
  const int*   rows  = (const int*)d_in[3];
  const int*   cols  = (const int*)d_in[4];
  const float* vals  = (const float*)d_in[5];
  const float* eu    = (const float*)d_in[6];
  const float* ei    = (const float*)d_in[7];
  const float* exu   = (const float*)d_in[8];
  const float* ei1   = (const float*)d_in[9];
  const float* ei0   = (const float*)d_in[10];
  const float* wu    = (const float*)d_in[11];
  const float* wi    = (const float*)d_in[12];
  float* out = (float*)d_out;

  const long long ND = (long long)N_CNT * D_DIM;   // 9.6M floats = 38.4 MB
  float* buf0 = (float*)d_ws;
  float* buf1 = buf0 + ND;
  float* acc  = buf1 + ND;                          // total 115.2 MB of d_ws

  const int bs = 256;
  const int gND = (int)((ND + bs - 1) / bs);

  lg_init_kernel<<<gND, bs, 0, stream>>>(eu, ei, buf0, acc);

  float* x  = buf0;
  float* xn = buf1;
  for (int l = 0; l < NLAYERS; ++l) {
    lg_zero_kernel<<<gND, bs, 0, stream>>>(xn, ND);
    lg_spmm_kernel<<<(NNZ_CNT + 7) / 8, 256, 0, stream>>>(rows, cols, vals, x, xn);
    lg_accadd_kernel<<<gND, bs, 0, stream>>>(acc, xn, ND);
    float* t = x; x = xn; xn = t;
  }

  // xn is now the stale ping-pong buffer -> reuse for the dense linear outputs
  float* ulin = xn;
  float* ilin = xn + (long long)B_CNT * D_DIM;

  lg_gemm_wmma_kernel<<<B_CNT / 16, 32, 0, stream>>>(users, items, acc, wu, wi, ulin, ilin);
  lg_gamma_kernel<<<B_CNT / 8, 256, 0, stream>>>(users, items, xij, ulin, ilin, exu, ei1, ei0, out);
}